// RNNCellModel_20066087207208
// MI455X (gfx1250) — compile-verified
//
#include <hip/hip_runtime.h>

typedef __attribute__((ext_vector_type(2))) float v2f;
typedef __attribute__((ext_vector_type(8))) float v8f;

#define SEQ   10
#define BATCH 2097152
#define IN    3
#define HID   5

// Cheap, branch-free tanh. gfx1250 has a native V_TANH_F32 trans op; use the
// builtin when declared, else the exp identity (saturates correctly at +/-1
// because exp overflows to +inf / underflows to 0).
__device__ __forceinline__ float fast_tanh(float v) {
#if __has_builtin(__builtin_amdgcn_tanhf)
    return __builtin_amdgcn_tanhf(v);
#else
    float e = __expf(2.0f * v);
    return 1.0f - 2.0f / (e + 1.0f);
#endif
}

// One wave handles TWO 16-batch tiles (32 batch elements).
//   A (16x4 f32): weights, M = hidden row (0..4 valid), K striped:
//                 lanes 0-15 hold K=0 (v0) / K=1 (v1); lanes 16-31 K=2 / K=3.
//   B (4x16 f32): data, N = batch lane (0..15), K striped the same way.
//   C/D (16x16 f32, 8 VGPRs): VGPR r = row M=r (lanes 0-15) / M=r+8 (lanes 16-31).
// Only rows 0..4 / lanes 0..15 of D are meaningful; the rest is zero padding.
__global__ __launch_bounds__(256) void rnncell_wmma_kernel(
    const float* __restrict__ x,    const float* __restrict__ hx0,
    const float* __restrict__ W_ih, const float* __restrict__ W_hh,
    const float* __restrict__ b_ih, const float* __restrict__ b_hh,
    float* __restrict__ out)
{
    const int  lane = threadIdx.x & 31;
    const int  wave = threadIdx.x >> 5;
    const long long b0 = ((long long)blockIdx.x * 8 + wave) * 32;
    const int  nn = lane & 15;
    const bool lo = lane < 16;
    const int  m  = nn;

    // ---- Loop-invariant A operands (weights in WMMA 16x4 f32 A layout) ----
    v2f a_ih;
    a_ih.x = (m < HID) ? (lo ? W_ih[m * IN + 0] : W_ih[m * IN + 2]) : 0.f; // K=0 / K=2
    a_ih.y = (m < HID && lo) ? W_ih[m * IN + 1] : 0.f;                     // K=1 / K=3(pad)

    v2f a_hh_lo, a_hh_hi;                                                  // K split 0..3 / 4..7
    a_hh_lo.x = (m < HID) ? (lo ? W_hh[m * HID + 0] : W_hh[m * HID + 2]) : 0.f;
    a_hh_lo.y = (m < HID) ? (lo ? W_hh[m * HID + 1] : W_hh[m * HID + 3]) : 0.f;
    a_hh_hi.x = (m < HID && lo) ? W_hh[m * HID + 4] : 0.f;
    a_hh_hi.y = 0.f;

    // ---- Bias as loop-invariant C matrix: C[m][n] = b_ih[m] + b_hh[m] ----
    v8f cbias;
#pragma unroll
    for (int r = 0; r < 8; ++r) {
        float v = 0.f;
        if (lo && r < HID) v = b_ih[r] + b_hh[r];   // rows 5..15 (incl. all hi lanes) -> 0
        cbias[r] = v;
    }

    // ---- Per-tile element ids and streaming pointers (advance by constants) ----
    const long long e0 = b0 + nn;         // tile 0
    const long long e1 = b0 + 16 + nn;    // tile 1
    const float* px0a = x + e0 * IN + (lo ? 0 : 2);   // B.x source (K=0 / K=2)
    const float* px0b = x + e0 * IN + 1;              // B.y source (K=1); hi lanes masked
    const float* px1a = x + e1 * IN + (lo ? 0 : 2);
    const float* px1b = x + e1 * IN + 1;
    float* po0 = out + e0 * HID;
    float* po1 = out + e1 * HID;
    float* pf0 = out + (long long)SEQ * BATCH * HID + e0 * HID;
    float* pf1 = out + (long long)SEQ * BATCH * HID + e1 * HID;

    // ---- Initial hidden state in B layout (K x N) for both tiles ----
    v2f bh0_lo, bh0_hi, bh1_lo, bh1_hi;
    {
        const float* h0p = hx0 + e0 * HID;
        const float* h1p = hx0 + e1 * HID;
        bh0_lo.x = h0p[lo ? 0 : 2];  bh0_lo.y = h0p[lo ? 1 : 3];
        float t0 = h0p[4];           bh0_hi.x = lo ? t0 : 0.f;  bh0_hi.y = 0.f;
        bh1_lo.x = h1p[lo ? 0 : 2];  bh1_lo.y = h1p[lo ? 1 : 3];
        float t1 = h1p[4];           bh1_hi.x = lo ? t1 : 0.f;  bh1_hi.y = 0.f;
    }

#pragma unroll
    for (int t = 0; t < SEQ; ++t) {
        // x tiles in B layout (unconditional loads -> always in-bounds, then mask)
        v2f bx0, bx1;
        bx0.x = px0a[0];  { float v = px0b[0]; bx0.y = lo ? v : 0.f; }
        bx1.x = px1a[0];  { float v = px1b[0]; bx1.y = lo ? v : 0.f; }
        px0a += (long long)BATCH * IN;  px0b += (long long)BATCH * IN;
        px1a += (long long)BATCH * IN;  px1b += (long long)BATCH * IN;

        // D = W_ih*X + bias; D += W_hh[:,0:4]*H[0:4]; D += W_hh[:,4]*H[4]
        v8f d0, d1;
        d0 = __builtin_amdgcn_wmma_f32_16x16x4_f32(false, a_ih,    false, bx0,
                                                   (short)0, cbias, false, false);
        d1 = __builtin_amdgcn_wmma_f32_16x16x4_f32(false, a_ih,    false, bx1,
                                                   (short)0, cbias, false, false);
        d0 = __builtin_amdgcn_wmma_f32_16x16x4_f32(false, a_hh_lo, false, bh0_lo,
                                                   (short)0, d0,    false, false);
        d1 = __builtin_amdgcn_wmma_f32_16x16x4_f32(false, a_hh_lo, false, bh1_lo,
                                                   (short)0, d1,    false, false);
        d0 = __builtin_amdgcn_wmma_f32_16x16x4_f32(false, a_hh_hi, false, bh0_hi,
                                                   (short)0, d0,    false, false);
        d1 = __builtin_amdgcn_wmma_f32_16x16x4_f32(false, a_hh_hi, false, bh1_hi,
                                                   (short)0, d1,    false, false);

        // h_new = tanh(D); valid rows 0..4 in lanes 0..15
        const float h00 = fast_tanh(d0[0]), h01 = fast_tanh(d0[1]);
        const float h02 = fast_tanh(d0[2]), h03 = fast_tanh(d0[3]);
        const float h04 = fast_tanh(d0[4]);
        const float h10 = fast_tanh(d1[0]), h11 = fast_tanh(d1[1]);
        const float h12 = fast_tanh(d1[2]), h13 = fast_tanh(d1[3]);
        const float h14 = fast_tanh(d1[4]);

        // ys[t][b][:]
        if (lo) {
            po0[0] = h00; po0[1] = h01; po0[2] = h02; po0[3] = h03; po0[4] = h04;
            po1[0] = h10; po1[1] = h11; po1[2] = h12; po1[3] = h13; po1[4] = h14;
        }
        po0 += (long long)BATCH * HID;
        po1 += (long long)BATCH * HID;

        // Rebuild B-layout hidden operand: rows K=2,3 move to lanes 16-31.
        const float s02 = __shfl(h02, nn, 32);
        const float s03 = __shfl(h03, nn, 32);
        const float s12 = __shfl(h12, nn, 32);
        const float s13 = __shfl(h13, nn, 32);
        bh0_lo.x = lo ? h00 : s02;   bh0_lo.y = lo ? h01 : s03;
        bh0_hi.x = lo ? h04 : 0.f;   bh0_hi.y = 0.f;
        bh1_lo.x = lo ? h10 : s12;   bh1_lo.y = lo ? h11 : s13;
        bh1_hi.x = lo ? h14 : 0.f;   bh1_hi.y = 0.f;

        // h_final
        if (t == SEQ - 1 && lo) {
            pf0[0] = h00; pf0[1] = h01; pf0[2] = h02; pf0[3] = h03; pf0[4] = h04;
            pf1[0] = h10; pf1[1] = h11; pf1[2] = h12; pf1[3] = h13; pf1[4] = h14;
        }
    }
}

extern "C" void kernel_launch(void* const* d_in, const int* in_sizes, int n_in,
                              void* d_out, int out_size, void* d_ws, size_t ws_size,
                              hipStream_t stream) {
    (void)in_sizes; (void)n_in; (void)d_ws; (void)ws_size; (void)out_size;
    const float* x    = (const float*)d_in[0];
    const float* hx0  = (const float*)d_in[1];
    const float* W_ih = (const float*)d_in[2];
    const float* W_hh = (const float*)d_in[3];
    const float* b_ih = (const float*)d_in[4];
    const float* b_hh = (const float*)d_in[5];
    float* out = (float*)d_out;

    // 32 batch elements per wave, 8 waves (256 threads) per block.
    const int blocks = BATCH / (32 * 8);   // 8192
    rnncell_wmma_kernel<<<blocks, 256, 0, stream>>>(x, hx0, W_ih, W_hh, b_ih, b_hh, out);
}